// LocallyEnhancedAttention_76647986364707
// MI455X (gfx1250) — compile-verified
//
#include <hip/hip_runtime.h>
#include <math.h>

// ---------------------------------------------------------------------------
// LocallyEnhancedAttention for MI455X (gfx1250, wave32, WMMA)
// B=16, N=3136 (56x56), C=256, HEADS=8, hd=32, SR=7 -> pooled 8x8, N_=64
// ---------------------------------------------------------------------------

typedef __attribute__((ext_vector_type(2))) float v2f;
typedef __attribute__((ext_vector_type(8))) float v8f;

#define B_      16
#define N_TOK   3136
#define C_      256
#define H_IMG   56
#define NP      64      // pooled tokens (8x8)
#define HEADS_  8
#define HD      32
#define EPSBN   1e-5f

__device__ __forceinline__ float gelu_exact(float x) {
  return 0.5f * x * (1.0f + erff(x * 0.70710678118654752f));
}
__device__ __forceinline__ int clampi(int v, int lo, int hi) {
  return v < lo ? lo : (v > hi ? hi : v);
}

// ---------------------------------------------------------------------------
// GEMM: OUT[m][n] = sum_k A[m][k] * W[n][k] + bias[n]   (torch Linear)
// K = C_ = 256.  Block = 256 threads (8 wave32), tile = 128 rows x 16 cols.
// Weight panel staged in LDS, K-pair interleaved so B fragments are b64 loads.
// Uses native fp32 WMMA 16x16x4.
// ---------------------------------------------------------------------------
__global__ __launch_bounds__(256) void gemm_xwt_wmma(
    const float* __restrict__ A, const float* __restrict__ W,
    const float* __restrict__ bias, float* __restrict__ OUT) {
  __shared__ float Wl[C_ * 16]; // layout: [(k>>1)*16 + n]*2 + (k&1)
  const int tid   = threadIdx.x;
  const int wave  = tid >> 5;
  const int lane  = tid & 31;
  const int laneN = lane & 15;
  const int koff  = (lane >> 4) << 1;          // 0 or 2
  const long rowBase = (long)blockIdx.x * 128;
  const int  colBase = blockIdx.y * 16;

  // stage 16-col weight panel (coalesced 8B per thread)
  for (int i = tid; i < (C_ / 2) * 16; i += 256) {
    const int k2 = i & 127;          // k/2 (fastest -> coalesced along K)
    const int n  = i >> 7;
    const v2f wv = *(const v2f*)(W + (size_t)(colBase + n) * C_ + (k2 << 1));
    *(v2f*)&Wl[(k2 * 16 + n) * 2] = wv;
  }
  __syncthreads();

  const float* __restrict__ arow = A + (rowBase + wave * 16 + laneN) * (long)C_;
  v8f c = {0.f, 0.f, 0.f, 0.f, 0.f, 0.f, 0.f, 0.f};
#pragma unroll 8
  for (int ks = 0; ks < C_; ks += 4) {
    const v2f a = *(const v2f*)(arow + ks + koff);
    const v2f b = *(const v2f*)(&Wl[(((ks + koff) >> 1) * 16 + laneN) * 2]);
    c = __builtin_amdgcn_wmma_f32_16x16x4_f32(false, a, false, b, (short)0, c,
                                              false, false);
  }

  const int  n  = colBase + laneN;
  const float bv = bias ? bias[n] : 0.0f;
  const long r0  = rowBase + wave * 16 + ((lane >> 4) << 3);
#pragma unroll
  for (int r = 0; r < 8; ++r) OUT[(r0 + r) * (long)C_ + n] = c[r] + bv;
}

// ---------------------------------------------------------------------------
// attn_small[bh][i][j] = sum_d q[bh][i][d] * k[bh][j][d]   (M=N=64, K=32)
// one wave per 16x16 tile, 8 waves per block.
// ---------------------------------------------------------------------------
__global__ __launch_bounds__(256) void qk_wmma_kernel(
    const float* __restrict__ qp, const float* __restrict__ kp,
    float* __restrict__ attnS) {
  const int tile  = blockIdx.x * 8 + (threadIdx.x >> 5);  // 0..2047
  const int lane  = threadIdx.x & 31;
  const int laneN = lane & 15;
  const int koff  = (lane >> 4) << 1;
  const int tj = tile & 3;
  const int ti = (tile >> 2) & 3;
  const int bh = tile >> 4;  // 0..127
  const float* __restrict__ qrow = qp + ((size_t)bh * NP + ti * 16 + laneN) * HD;
  const float* __restrict__ krow = kp + ((size_t)bh * NP + tj * 16 + laneN) * HD;
  v8f c = {0.f, 0.f, 0.f, 0.f, 0.f, 0.f, 0.f, 0.f};
#pragma unroll
  for (int ks = 0; ks < HD; ks += 4) {
    const v2f a = *(const v2f*)(qrow + ks + koff);
    const v2f b = *(const v2f*)(krow + ks + koff);
    c = __builtin_amdgcn_wmma_f32_16x16x4_f32(false, a, false, b, (short)0, c,
                                              false, false);
  }
  float* __restrict__ orow = attnS + (size_t)bh * (NP * NP);
  const int r0 = ti * 16 + ((lane >> 4) << 3);
#pragma unroll
  for (int r = 0; r < 8; ++r) orow[(r0 + r) * NP + tj * 16 + laneN] = c[r];
}

// ---------------------------------------------------------------------------
// Depthwise 3x3 conv + eval BN, token-major [B,N,C] layout (channel fastest).
// Optionally applies exact GELU to the input values (for the s2 branch).
// ---------------------------------------------------------------------------
template <bool GELU_IN>
__global__ __launch_bounds__(256) void dwconv_bn_kernel(
    const float* __restrict__ in, const float* __restrict__ w,
    const float* __restrict__ gg, const float* __restrict__ bb,
    const float* __restrict__ mm, const float* __restrict__ vv,
    float* __restrict__ out) {
  const size_t idx = (size_t)blockIdx.x * 256 + threadIdx.x;
  const int c = (int)(idx & (C_ - 1));
  const size_t bn = idx >> 8;
  const int n = (int)(bn % N_TOK);
  const int b = (int)(bn / N_TOK);
  const int y = n / H_IMG, x = n - y * H_IMG;
  const float* __restrict__ base = in + (size_t)b * N_TOK * C_;
  float acc = 0.f;
#pragma unroll
  for (int ky = 0; ky < 3; ++ky) {
    const int yy = y + ky - 1;
    if (yy < 0 || yy >= H_IMG) continue;
#pragma unroll
    for (int kx = 0; kx < 3; ++kx) {
      const int xx = x + kx - 1;
      if (xx < 0 || xx >= H_IMG) continue;
      float val = base[(size_t)(yy * H_IMG + xx) * C_ + c];
      if (GELU_IN) val = gelu_exact(val);
      acc += val * w[c * 9 + ky * 3 + kx];
    }
  }
  const float sc = gg[c] * rsqrtf(vv[c] + EPSBN);
  out[idx] = acc * sc + (bb[c] - mm[c] * sc);
}

// ---------------------------------------------------------------------------
// 7x7 avg-pool from [B,N,C] into head-major pooled layout [B,H,N_,hd]
// ---------------------------------------------------------------------------
__global__ __launch_bounds__(256) void avgpool_kernel(
    const float* __restrict__ s, float* __restrict__ outp) {
  const int idx = blockIdx.x * 256 + threadIdx.x;  // B*NP*C
  const int c  = idx & (C_ - 1);
  const int t  = idx >> 8;
  const int np = t & (NP - 1);
  const int b  = t >> 6;
  const int yy = np >> 3, xx = np & 7;
  const float* __restrict__ base =
      s + ((size_t)b * N_TOK + (size_t)(yy * 7) * H_IMG + xx * 7) * C_ + c;
  float acc = 0.f;
#pragma unroll
  for (int dy = 0; dy < 7; ++dy)
#pragma unroll
    for (int dx = 0; dx < 7; ++dx)
      acc += base[(size_t)(dy * H_IMG + dx) * C_];
  const int hID = c >> 5, d = c & 31;
  outp[(((size_t)b * HEADS_ + hID) * NP + np) * HD + d] = acc * (1.0f / 49.0f);
}

// ---------------------------------------------------------------------------
// Fused: bilinear upsample (8x8 -> 56x56) of one attention row, *scale,
// softmax over all 3136 keys (row kept in LDS), then P @ V for 32 head dims.
// One 256-thread block per (b, head, pooled-query). Avoids materializing the
// 102 MB upsampled attention tensor (~300 MB of HBM traffic saved).
// ---------------------------------------------------------------------------
__global__ __launch_bounds__(256) void attn_row_kernel(
    const float* __restrict__ attnS, const float* __restrict__ v,
    float* __restrict__ osm) {
  __shared__ float amap[NP];
  __shared__ float p[N_TOK];
  __shared__ float red[256];
  const int t   = threadIdx.x;
  const int i   = blockIdx.x & (NP - 1);
  const int hID = (blockIdx.x >> 6) & (HEADS_ - 1);
  const int b   = blockIdx.x >> 9;
  if (t < NP) amap[t] = attnS[(((size_t)(b * HEADS_ + hID)) * NP + i) * NP + t];
  __syncthreads();

  const float scale = 0.0625f;  // C^-0.5
  float lmax = -3.0e38f;
  for (int m = t; m < N_TOK; m += 256) {
    const int y = m / H_IMG, x = m - y * H_IMG;
    const float fy = (y + 0.5f) * (1.0f / 7.0f) - 0.5f;
    const float fx = (x + 0.5f) * (1.0f / 7.0f) - 0.5f;
    const int y0 = (int)floorf(fy);
    const int x0 = (int)floorf(fx);
    const float wy = fy - (float)y0, wx = fx - (float)x0;
    const int iy0 = clampi(y0, 0, 7), iy1 = clampi(y0 + 1, 0, 7);
    const int ix0 = clampi(x0, 0, 7), ix1 = clampi(x0 + 1, 0, 7);
    const float v00 = amap[iy0 * 8 + ix0], v01 = amap[iy0 * 8 + ix1];
    const float v10 = amap[iy1 * 8 + ix0], v11 = amap[iy1 * 8 + ix1];
    const float val = ((1.f - wy) * ((1.f - wx) * v00 + wx * v01) +
                       wy * ((1.f - wx) * v10 + wx * v11)) * scale;
    p[m] = val;
    lmax = fmaxf(lmax, val);
  }
  red[t] = lmax;
  __syncthreads();
  for (int s = 128; s > 0; s >>= 1) {
    if (t < s) red[t] = fmaxf(red[t], red[t + s]);
    __syncthreads();
  }
  const float rmax = red[0];
  __syncthreads();

  float lsum = 0.f;
  for (int m = t; m < N_TOK; m += 256) {
    const float e = __expf(p[m] - rmax);
    p[m] = e;
    lsum += e;
  }
  red[t] = lsum;
  __syncthreads();
  for (int s = 128; s > 0; s >>= 1) {
    if (t < s) red[t] += red[t + s];
    __syncthreads();
  }
  const float rinv = 1.0f / red[0];
  __syncthreads();

  // P @ V : 8 m-groups x 32 head dims; v reads are 128B-coalesced per group
  const int g = t >> 5, d = t & 31;
  const float* __restrict__ vcol = v + (size_t)b * N_TOK * C_ + hID * HD + d;
  float acc = 0.f;
  for (int m = g; m < N_TOK; m += 8) acc += p[m] * vcol[(size_t)m * C_];
  red[t] = acc;
  __syncthreads();
  if (t < 32) {
    float tot = 0.f;
#pragma unroll
    for (int gg2 = 0; gg2 < 8; ++gg2) tot += red[gg2 * 32 + t];
    osm[((size_t)b * NP + i) * C_ + hID * HD + t] = tot * rinv;
  }
}

// ---------------------------------------------------------------------------
// Fused: bilinear upsample of o (8x8 -> 56x56) computed on the fly per tap,
// depthwise 3x3 conv (SAME zero-pad) + BN3, plus GELU(s2) skip connection.
// osm is [B, N_=64, C] (channel fastest, coalesced reads, ~1 MB -> L2 hot).
// ---------------------------------------------------------------------------
__global__ __launch_bounds__(256) void upconv_bn_add_kernel(
    const float* __restrict__ osm, const float* __restrict__ w,
    const float* __restrict__ gg, const float* __restrict__ bb,
    const float* __restrict__ mm, const float* __restrict__ vv,
    const float* __restrict__ s2, float* __restrict__ out) {
  const size_t idx = (size_t)blockIdx.x * 256 + threadIdx.x;
  const int c = (int)(idx & (C_ - 1));
  const size_t bn = idx >> 8;
  const int n = (int)(bn % N_TOK);
  const int b = (int)(bn / N_TOK);
  const int y = n / H_IMG, x = n - y * H_IMG;
  const float* __restrict__ base = osm + (size_t)b * NP * C_ + c;
  float acc = 0.f;
#pragma unroll
  for (int ky = 0; ky < 3; ++ky) {
    const int yy = y + ky - 1;
    if (yy < 0 || yy >= H_IMG) continue;
    const float fy = (yy + 0.5f) * (1.0f / 7.0f) - 0.5f;
    const int y0 = (int)floorf(fy);
    const float wy = fy - (float)y0;
    const int iy0 = clampi(y0, 0, 7), iy1 = clampi(y0 + 1, 0, 7);
#pragma unroll
    for (int kx = 0; kx < 3; ++kx) {
      const int xx = x + kx - 1;
      if (xx < 0 || xx >= H_IMG) continue;
      const float fx = (xx + 0.5f) * (1.0f / 7.0f) - 0.5f;
      const int x0 = (int)floorf(fx);
      const float wx = fx - (float)x0;
      const int ix0 = clampi(x0, 0, 7), ix1 = clampi(x0 + 1, 0, 7);
      const float v00 = base[(size_t)(iy0 * 8 + ix0) * C_];
      const float v01 = base[(size_t)(iy0 * 8 + ix1) * C_];
      const float v10 = base[(size_t)(iy1 * 8 + ix0) * C_];
      const float v11 = base[(size_t)(iy1 * 8 + ix1) * C_];
      const float valu = (1.f - wy) * ((1.f - wx) * v00 + wx * v01) +
                         wy * ((1.f - wx) * v10 + wx * v11);
      acc += valu * w[c * 9 + ky * 3 + kx];
    }
  }
  const float sc = gg[c] * rsqrtf(vv[c] + EPSBN);
  const float o  = acc * sc + (bb[c] - mm[c] * sc);
  out[idx] = o + gelu_exact(s2[idx]);
}

// ---------------------------------------------------------------------------
extern "C" void kernel_launch(void* const* d_in, const int* in_sizes, int n_in,
                              void* d_out, int out_size, void* d_ws,
                              size_t ws_size, hipStream_t stream) {
  const float* x       = (const float*)d_in[0];
  const float* Wv      = (const float*)d_in[1];
  const float* conv1_w = (const float*)d_in[2];
  const float* bn1_g   = (const float*)d_in[3];
  const float* bn1_b   = (const float*)d_in[4];
  const float* bn1_m   = (const float*)d_in[5];
  const float* bn1_v   = (const float*)d_in[6];
  const float* conv2_w = (const float*)d_in[7];
  const float* bn2_g   = (const float*)d_in[8];
  const float* bn2_b   = (const float*)d_in[9];
  const float* bn2_m   = (const float*)d_in[10];
  const float* bn2_v   = (const float*)d_in[11];
  const float* vup_w   = (const float*)d_in[12];
  const float* bn3_g   = (const float*)d_in[13];
  const float* bn3_b   = (const float*)d_in[14];
  const float* bn3_m   = (const float*)d_in[15];
  const float* bn3_v   = (const float*)d_in[16];
  const float* Wp      = (const float*)d_in[17];
  const float* bp      = (const float*)d_in[18];
  float* out = (float*)d_out;

  // workspace layout (floats): 3 x [B,N,C] big buffers + small buffers
  const size_t BNC  = (size_t)B_ * N_TOK * C_;       // 12,845,056
  const size_t PBHD = (size_t)B_ * HEADS_ * NP * HD; // 262,144
  float* ws    = (float*)d_ws;
  float* v     = ws;                 // [B,N,C]
  float* s1    = ws + BNC;           // [B,N,C]  (reused as y_sum later)
  float* s2    = ws + 2 * BNC;       // [B,N,C]
  float* qp    = ws + 3 * BNC;       // [B,H,N_,hd]
  float* kp    = qp + PBHD;          // [B,H,N_,hd]
  float* attnS = kp + PBHD;          // [B,H,N_,N_] = 524288
  float* osm   = attnS + (size_t)B_ * HEADS_ * NP * NP;  // [B,N_,C] = 262144
  float* ysum  = s1;                 // s1 is dead by the time we write this

  const dim3 gemmGrid(392, 16);  // M=50176/128, N=256/16
  const int  nBNC = (int)(BNC / 256);            // 50176 blocks
  const int  nPool = (B_ * NP * C_) / 256;       // 1024 blocks

  // 1) v = x @ Wv^T                         (WMMA f32 GEMM)
  gemm_xwt_wmma<<<gemmGrid, 256, 0, stream>>>(x, Wv, nullptr, v);
  // 2) s1 = BN1(dwconv1(v))
  dwconv_bn_kernel<false><<<nBNC, 256, 0, stream>>>(v, conv1_w, bn1_g, bn1_b,
                                                    bn1_m, bn1_v, s1);
  // 3) q = avgpool7(s1) -> [B,H,N_,hd]
  avgpool_kernel<<<nPool, 256, 0, stream>>>(s1, qp);
  // 4) s2 = BN2(dwconv2(gelu(s1)))
  dwconv_bn_kernel<true><<<nBNC, 256, 0, stream>>>(s1, conv2_w, bn2_g, bn2_b,
                                                   bn2_m, bn2_v, s2);
  // 5) k = avgpool7(s2) -> [B,H,N_,hd]
  avgpool_kernel<<<nPool, 256, 0, stream>>>(s2, kp);
  // 6) attn_small = q @ k^T                 (WMMA f32 GEMM)
  qk_wmma_kernel<<<256, 256, 0, stream>>>(qp, kp, attnS);
  // 7) fused upsample + softmax + P@V -> o  [B,N_,C]
  attn_row_kernel<<<B_ * HEADS_ * NP, 256, 0, stream>>>(attnS, v, osm);
  // 8) y = BN3(dwconv3(upsample(o))) + gelu(s2)
  upconv_bn_add_kernel<<<nBNC, 256, 0, stream>>>(osm, vup_w, bn3_g, bn3_b,
                                                 bn3_m, bn3_v, s2, ysum);
  // 9) out = y @ Wp^T + bp                  (WMMA f32 GEMM)
  gemm_xwt_wmma<<<gemmGrid, 256, 0, stream>>>(ysum, Wp, bp, out);
}